// StatisticAttentionBlock_33904471835216
// MI455X (gfx1250) — compile-verified
//
#include <hip/hip_runtime.h>
#include <hip/hip_bf16.h>
#include <math.h>

// Problem constants (StatisticAttentionBlock)
#define B_   8
#define CIN_ 768
#define CAD_ 192
#define N_   12544          // T*H*W = 16*28*28
#define BN_EPS_ 1e-5f

typedef float v2f __attribute__((ext_vector_type(2)));
typedef float v8f __attribute__((ext_vector_type(8)));

// ---------------------------------------------------------------------------
// Block-wide sum reduction (256 threads), result broadcast to all threads.
// ---------------------------------------------------------------------------
__device__ __forceinline__ float block_reduce_sum256(float v, float* red) {
    const int t = threadIdx.x;
    red[t] = v;
    __syncthreads();
    for (int s = 128; s > 0; s >>= 1) {
        if (t < s) red[t] += red[t + s];
        __syncthreads();
    }
    float r = red[0];
    __syncthreads();
    return r;
}

// ---------------------------------------------------------------------------
// Kernel 1: down-projection GEMM  z[b,a,n] = sum_c w_down[a,c]*x[b,c,n] + b_down[a]
// Grid: (N/32, B), block: 384 threads = 12 waves; wave w owns rows [16w,16w+16)
// of CAD (12*16 = 192) and TWO 16-wide n-subtiles (A fragment reused 2x).
// x chunks (32 x 32 = 4 KB) staged via LDS, shared by all 12 waves.
// ---------------------------------------------------------------------------
__global__ __launch_bounds__(384) void sab_down_gemm(
        const float* __restrict__ x, const float* __restrict__ w_down,
        const float* __restrict__ b_down, float* __restrict__ z) {
    __shared__ float xs[32 * 32];

    const int n0   = blockIdx.x * 32;
    const int b    = blockIdx.y;
    const int t    = threadIdx.x;
    const int wave = t >> 5;
    const int lane = t & 31;
    const int half = lane >> 4;   // 0 or 1
    const int ln   = lane & 15;   // 0..15
    const int kbase = half * 2;
    const int a0   = wave * 16;

    const float* xb = x + (size_t)b * CIN_ * N_ + n0;

    v8f acc0 = {}, acc1 = {};
    for (int c0 = 0; c0 < CIN_; c0 += 32) {
        // cooperative load of 32x32 x-chunk (coalesced 128B rows)
        for (int idx = t; idx < 32 * 32; idx += 384) {
            const int c = idx >> 5, n = idx & 31;
            xs[idx] = xb[(size_t)(c0 + c) * N_ + n];
        }
        __syncthreads();
        #pragma unroll
        for (int kk = 0; kk < 8; ++kk) {
            const int k = kk * 4;
            // A = w_down[a0+ln][c0+k+kbase .. +1]  (row-major, 8B aligned)
            const v2f av = *(const v2f*)(w_down + (size_t)(a0 + ln) * CIN_ + (c0 + k + kbase));
            v2f bv0, bv1;
            bv0.x = xs[(k + kbase)     * 32 + ln];
            bv0.y = xs[(k + kbase + 1) * 32 + ln];
            bv1.x = xs[(k + kbase)     * 32 + 16 + ln];
            bv1.y = xs[(k + kbase + 1) * 32 + 16 + ln];
            acc0 = __builtin_amdgcn_wmma_f32_16x16x4_f32(
                    false, av, false, bv0, (short)0, acc0, false, false);
            acc1 = __builtin_amdgcn_wmma_f32_16x16x4_f32(
                    false, av, false, bv1, (short)0, acc1, false, false);
        }
        __syncthreads();
    }

    float* zb = z + (size_t)b * CAD_ * N_ + n0;
    #pragma unroll
    for (int r = 0; r < 8; ++r) {
        const int m = a0 + r + half * 8;
        const float bias = b_down[m];
        zb[(size_t)m * N_ + ln]      = acc0[r] + bias;
        zb[(size_t)m * N_ + 16 + ln] = acc1[r] + bias;
    }
}

// ---------------------------------------------------------------------------
// Kernel 2: per-(b,a) row moments over n: [mean, std, m4/std^4, m5/std^5, m6/std^6]
// Grid: B*CAD blocks of 256 threads. z is L2-resident (77MB < 192MB L2).
// ---------------------------------------------------------------------------
__global__ __launch_bounds__(256) void sab_moments(
        const float* __restrict__ z, float* __restrict__ mv) {
    __shared__ float red[256];
    const int row = blockIdx.x;                 // b*CAD + a
    const float* p = z + (size_t)row * N_;
    const int t = threadIdx.x;

    float s1 = 0.f, s2 = 0.f;
    for (int i = t; i < N_; i += 256) {
        const float v = p[i];
        s1 += v; s2 += v * v;
    }
    const float tot1 = block_reduce_sum256(s1, red);
    const float tot2 = block_reduce_sum256(s2, red);
    const float invn = 1.0f / (float)N_;
    const float mean = tot1 * invn;
    const float var  = fmaxf(tot2 * invn - mean * mean, 0.f);
    const float stdv = sqrtf(var);

    float s4 = 0.f, s5 = 0.f, s6 = 0.f;
    for (int i = t; i < N_; i += 256) {
        const float d  = p[i] - mean;
        const float d2 = d * d;
        s4 += d2 * d2;
        s5 += d2 * d2 * d;
        s6 += d2 * d2 * d2;
    }
    const float t4 = block_reduce_sum256(s4, red) * invn;
    const float t5 = block_reduce_sum256(s5, red) * invn;
    const float t6 = block_reduce_sum256(s6, red) * invn;

    if (t == 0) {
        float* o = mv + (size_t)row * 5;
        const float v2 = var * var;             // std^4
        o[0] = mean;
        o[1] = stdv;
        o[2] = t4 / v2;                         // m4 / std^4
        o[3] = t5 / (v2 * stdv);                // m5 / std^5
        o[4] = t6 / (v2 * var);                 // m6 / std^6
    }
}

// ---------------------------------------------------------------------------
// Kernel 3: normalize moment vectors along channel dim (192) per (b, m).
// Grid: B blocks of 256 threads.
// ---------------------------------------------------------------------------
__global__ __launch_bounds__(256) void sab_mvn(
        const float* __restrict__ mv, float* __restrict__ mvn) {
    __shared__ float red[256];
    const int b = blockIdx.x;
    const int t = threadIdx.x;
    for (int m = 0; m < 5; ++m) {
        float v = 0.f;
        if (t < CAD_) v = mv[((size_t)b * CAD_ + t) * 5 + m];
        const float ss   = block_reduce_sum256(v * v, red);
        const float norm = fmaxf(sqrtf(ss), 1e-12f);
        if (t < CAD_) mvn[((size_t)b * CAD_ + t) * 5 + m] = v / norm;
        __syncthreads();
    }
}

// ---------------------------------------------------------------------------
// Kernel 4: fused sta_distribute + up-GEMM + BN(eval) + residual ReLU.
// Grid: (N/32, B), block: 256 threads = 8 waves; wave w handles 6 m-tiles
// (8*6*16 = 768 = CIN output rows) x TWO n-subtiles. Per 32-wide n-tile:
//   (a) load z tile (192x32) + mvn (192x5) to LDS
//   (b) lanes 0..31: f[n,m] = sum_c z[c,n]*mvn[c,m]; local softmax over m=5
//   (c) rebuild zd[c,n] = sum_m f[n,m]*mvn[c,m] in-place in LDS
//   (d) WMMA f32 16x16x4 over K=192, A fragment reused for both n-subtiles,
//       epilogue applies b_up, folded BN affine, residual x, ReLU.
// ---------------------------------------------------------------------------
__global__ __launch_bounds__(256) void sab_up_gemm(
        const float* __restrict__ z, const float* __restrict__ mvn,
        const float* __restrict__ w_up, const float* __restrict__ b_up,
        const float* __restrict__ gamma, const float* __restrict__ beta,
        const float* __restrict__ bmean, const float* __restrict__ bvar,
        const float* __restrict__ x, float* __restrict__ out) {
    __shared__ float zt[CAD_ * 32];   // z tile, then zd tile (24 KB)
    __shared__ float mvs[CAD_ * 5];   // normalized moment vectors
    __shared__ float fbuf[32 * 5];    // softmax weights per column

    const int n0   = blockIdx.x * 32;
    const int b    = blockIdx.y;
    const int t    = threadIdx.x;
    const int wave = t >> 5;
    const int lane = t & 31;
    const int half = lane >> 4;
    const int ln   = lane & 15;
    const int kbase = half * 2;

    const float* zb = z + (size_t)b * CAD_ * N_ + n0;
    for (int idx = t; idx < CAD_ * 32; idx += 256)
        zt[idx] = zb[(size_t)(idx >> 5) * N_ + (idx & 31)];
    const float* mvb = mvn + (size_t)b * CAD_ * 5;
    for (int idx = t; idx < CAD_ * 5; idx += 256)
        mvs[idx] = mvb[idx];
    __syncthreads();

    // (b) attention logits + softmax (m = 5, private per column)
    if (t < 32) {
        float s[5] = {0.f, 0.f, 0.f, 0.f, 0.f};
        for (int c = 0; c < CAD_; ++c) {
            const float zv = zt[c * 32 + t];
            #pragma unroll
            for (int m = 0; m < 5; ++m) s[m] += zv * mvs[c * 5 + m];
        }
        float mx = s[0];
        #pragma unroll
        for (int m = 1; m < 5; ++m) mx = fmaxf(mx, s[m]);
        float se = 0.f;
        #pragma unroll
        for (int m = 0; m < 5; ++m) { s[m] = expf(s[m] - mx); se += s[m]; }
        const float rs = 1.f / se;
        #pragma unroll
        for (int m = 0; m < 5; ++m) fbuf[t * 5 + m] = s[m] * rs;
    }
    __syncthreads();

    // (c) zd = f @ mvn^T, overwrite zt (reads only fbuf/mvs -> safe)
    for (int idx = t; idx < CAD_ * 32; idx += 256) {
        const int c = idx >> 5, n = idx & 31;
        float sv = 0.f;
        #pragma unroll
        for (int m = 0; m < 5; ++m) sv += fbuf[n * 5 + m] * mvs[c * 5 + m];
        zt[idx] = sv;
    }
    __syncthreads();

    // (d) up-GEMM via WMMA f32 + BN + residual ReLU
    const float* xb = x   + (size_t)b * CIN_ * N_ + n0;
    float*       ob = out + (size_t)b * CIN_ * N_ + n0;
    for (int tile = 0; tile < 6; ++tile) {
        const int o0 = (wave * 6 + tile) * 16;
        v8f acc0 = {}, acc1 = {};
        #pragma unroll 4
        for (int kk = 0; kk < CAD_ / 4; ++kk) {
            const int k = kk * 4;
            const v2f av = *(const v2f*)(w_up + (size_t)(o0 + ln) * CAD_ + (k + kbase));
            v2f bv0, bv1;
            bv0.x = zt[(k + kbase)     * 32 + ln];
            bv0.y = zt[(k + kbase + 1) * 32 + ln];
            bv1.x = zt[(k + kbase)     * 32 + 16 + ln];
            bv1.y = zt[(k + kbase + 1) * 32 + 16 + ln];
            acc0 = __builtin_amdgcn_wmma_f32_16x16x4_f32(
                    false, av, false, bv0, (short)0, acc0, false, false);
            acc1 = __builtin_amdgcn_wmma_f32_16x16x4_f32(
                    false, av, false, bv1, (short)0, acc1, false, false);
        }
        #pragma unroll
        for (int r = 0; r < 8; ++r) {
            const int o       = o0 + r + half * 8;
            const float inv   = gamma[o] * rsqrtf(bvar[o] + BN_EPS_);
            const float shift = beta[o] - bmean[o] * inv;
            const float bias  = b_up[o];
            float u0 = (acc0[r] + bias) * inv + shift;
            float u1 = (acc1[r] + bias) * inv + shift;
            u0 += xb[(size_t)o * N_ + ln];
            u1 += xb[(size_t)o * N_ + 16 + ln];
            ob[(size_t)o * N_ + ln]      = fmaxf(u0, 0.f);
            ob[(size_t)o * N_ + 16 + ln] = fmaxf(u1, 0.f);
        }
    }
}

// ---------------------------------------------------------------------------
// Host-side launcher
// ---------------------------------------------------------------------------
extern "C" void kernel_launch(void* const* d_in, const int* in_sizes, int n_in,
                              void* d_out, int out_size, void* d_ws, size_t ws_size,
                              hipStream_t stream) {
    (void)in_sizes; (void)n_in; (void)out_size; (void)ws_size;

    const float* x       = (const float*)d_in[0];
    const float* w_down  = (const float*)d_in[1];
    const float* b_down  = (const float*)d_in[2];
    const float* w_up    = (const float*)d_in[3];
    const float* b_up    = (const float*)d_in[4];
    const float* bn_g    = (const float*)d_in[5];
    const float* bn_b    = (const float*)d_in[6];
    const float* bn_m    = (const float*)d_in[7];
    const float* bn_v    = (const float*)d_in[8];
    float*       out     = (float*)d_out;

    // Workspace layout: z (77 MB, L2-resident) | mv | mvn
    float* z   = (float*)d_ws;
    float* mv  = z  + (size_t)B_ * CAD_ * N_;
    float* mvn = mv + (size_t)B_ * CAD_ * 5;

    sab_down_gemm<<<dim3(N_ / 32, B_), 384, 0, stream>>>(x, w_down, b_down, z);
    sab_moments  <<<B_ * CAD_,          256, 0, stream>>>(z, mv);
    sab_mvn      <<<B_,                 256, 0, stream>>>(mv, mvn);
    sab_up_gemm  <<<dim3(N_ / 32, B_), 256, 0, stream>>>(z, mvn, w_up, b_up,
                                                          bn_g, bn_b, bn_m, bn_v,
                                                          x, out);
}